// KeepTopK_49976239456757
// MI455X (gfx1250) — compile-verified
//
#include <hip/hip_runtime.h>
#include <stdint.h>

// KeepTopK: per 128-float row, keep top-16 values, zero the rest.
// Memory-bound streaming kernel: 268 MB traffic -> ~11.5us floor @ 23.3 TB/s.
// CDNA5 path: async global->LDS B128 loads (ASYNCcnt) with double buffering,
// wave32 shuffle/ballot top-k selection, non-temporal 128-bit stores.

#define WAVE_SZ          32
#define WAVES_PER_BLOCK  8
#define ROW              128
#define KTOP             16

typedef float v4f __attribute__((ext_vector_type(4)));

__global__ __launch_bounds__(WAVES_PER_BLOCK * WAVE_SZ)
void keep_topk_kernel(const float* __restrict__ x, float* __restrict__ out,
                      int rows, int stride_rows) {
    // Per-wave double buffer: 8 waves * 2 bufs * 512B = 8KB LDS (of 320KB/WGP).
    __shared__ __align__(16) float tile[WAVES_PER_BLOCK][2][ROW];

    const int lane = threadIdx.x & (WAVE_SZ - 1);
    const int wv   = threadIdx.x >> 5;
    const int row0 = blockIdx.x * WAVES_PER_BLOCK + wv;

    // LDS byte addresses of this lane's 16B chunk in each buffer
    // (low 32 bits of a generic pointer to __shared__ == LDS byte offset).
    const uint32_t lds0 = (uint32_t)(uintptr_t)(&tile[wv][0][lane * 4]);
    const uint32_t lds1 = (uint32_t)(uintptr_t)(&tile[wv][1][lane * 4]);

    // Prologue: kick off async load of first row into buffer 0.
    if (row0 < rows) {
        uint32_t goff = (uint32_t)row0 * (ROW * 4u) + (uint32_t)lane * 16u;
        asm volatile("global_load_async_to_lds_b128 %0, %1, %2"
                     :: "v"(lds0), "v"(goff), "s"(x) : "memory");
    }

    int buf = 0;
    for (int row = row0; row < rows; row += stride_rows) {
        const int nxt = row + stride_rows;
        if (nxt < rows) {
            // Prefetch next row into the other buffer, then wait for current
            // (async loads complete in order; <=1 outstanding == current done).
            uint32_t goff = (uint32_t)nxt * (ROW * 4u) + (uint32_t)lane * 16u;
            uint32_t l = buf ? lds0 : lds1;
            asm volatile("global_load_async_to_lds_b128 %0, %1, %2"
                         :: "v"(l), "v"(goff), "s"(x) : "memory");
            asm volatile("s_wait_asynccnt 0x1" ::: "memory");
        } else {
            asm volatile("s_wait_asynccnt 0x0" ::: "memory");
        }

        // Each lane owns elements [lane*4 .. lane*4+3] of the row (index-major
        // across lanes, so ballot-ctz picks the lowest global index on ties,
        // matching top_k tie-breaking).
        const v4f v = *reinterpret_cast<const v4f*>(&tile[wv][buf][lane * 4]);
        float w0 = v.x, w1 = v.y, w2 = v.z, w3 = v.w;
        unsigned kmask = 0u;

        #pragma unroll 1
        for (int it = 0; it < KTOP; ++it) {
            float lm = fmaxf(fmaxf(w0, w1), fmaxf(w2, w3));
            float m = lm;
            #pragma unroll
            for (int off = 16; off > 0; off >>= 1)
                m = fmaxf(m, __shfl_xor(m, off, WAVE_SZ));

            unsigned bal;
#if __has_builtin(__builtin_amdgcn_ballot_w32)
            bal = __builtin_amdgcn_ballot_w32(lm == m);
#else
            bal = (unsigned)__ballot(lm == m);
#endif
            const int first = __builtin_ctz(bal);
            if (lane == first) {
                if      (w0 == m) { kmask |= 1u; w0 = -__builtin_inff(); }
                else if (w1 == m) { kmask |= 2u; w1 = -__builtin_inff(); }
                else if (w2 == m) { kmask |= 4u; w2 = -__builtin_inff(); }
                else              { kmask |= 8u; w3 = -__builtin_inff(); }
            }
        }

        v4f o;
        o.x = (kmask & 1u) ? v.x : 0.0f;
        o.y = (kmask & 2u) ? v.y : 0.0f;
        o.z = (kmask & 4u) ? v.z : 0.0f;
        o.w = (kmask & 8u) ? v.w : 0.0f;
        // Output has zero reuse and exceeds L2: non-temporal 128-bit store.
        __builtin_nontemporal_store(
            o, reinterpret_cast<v4f*>(out + (size_t)row * ROW + lane * 4));

        buf ^= 1;
    }
}

extern "C" void kernel_launch(void* const* d_in, const int* in_sizes, int n_in,
                              void* d_out, int out_size, void* d_ws, size_t ws_size,
                              hipStream_t stream) {
    const float* x   = (const float*)d_in[0];
    float*       out = (float*)d_out;

    const int rows = in_sizes[0] / ROW;   // 262144
    int blocks = (rows + WAVES_PER_BLOCK - 1) / WAVES_PER_BLOCK;
    if (blocks > 4096) blocks = 4096;     // 32768 waves, ~8 rows each: keeps
                                          // the async double-buffer pipeline fed
    const int stride_rows = blocks * WAVES_PER_BLOCK;

    keep_topk_kernel<<<blocks, WAVES_PER_BLOCK * WAVE_SZ, 0, stream>>>(
        x, out, rows, stride_rows);
}